// CombinedLoss_65008624992506
// MI455X (gfx1250) — compile-verified
//
#include <hip/hip_runtime.h>
#include <cstdint>
#include <cstddef>

#define TW       128
#define THT      64
#define HALOX    4                   // padded x-halo (need 2) keeps 16B alignment
#define HALOY    2
#define LWP      (TW + 2 * HALOX)    // 136 floats per LDS row
#define LH       (THT + 2 * HALOY)   // 68 rows
#define NTHREADS 256
#define IMG      512
#define TILES_X  (IMG / TW)          // 4
#define TILES_Y  (IMG / THT)         // 8
#define NBLOCKS  (TILES_X * TILES_Y * 64)  // 2048
#define NACC     5

typedef uint32_t u4 __attribute__((ext_vector_type(4)));
typedef int      i4 __attribute__((ext_vector_type(4)));
typedef int      i8 __attribute__((ext_vector_type(8)));

struct RowAgg { float a, bmin, bmax, cmin, cmax; };

// Per-row aggregates for the radius-2 diamond stencil:
//  a = center (taps dy=+-2); bmin/bmax = 3-wide (dy=+-1); cmin/cmax = 5-wide (dy=0)
__device__ __forceinline__ RowAgg make_row(const float* s_row) {
  RowAgg r;
  float m2 = s_row[-2], m1 = s_row[-1], m0 = s_row[0], p1 = s_row[1], p2 = s_row[2];
  r.a = m0;
  r.bmin = fminf(fminf(m1, m0), p1);
  r.bmax = fmaxf(fmaxf(m1, m0), p1);
  r.cmin = fminf(fminf(m2, p2), r.bmin);
  r.cmax = fmaxf(fmaxf(m2, p2), r.bmax);
  return r;
}

__global__ __launch_bounds__(NTHREADS) void loss_partial_kernel(
    const float* __restrict__ inputs, const float* __restrict__ targets,
    float* __restrict__ partials) {
  __shared__ float s_t[LH * LWP];
  __shared__ float s_red[8][NACC];

  const int tid = threadIdx.x;
  const int blk = blockIdx.x;
  const int tx  = blk % TILES_X;
  const int ty  = (blk / TILES_X) % TILES_Y;
  const int b   = blk / (TILES_X * TILES_Y);
  const int x0  = tx * TW;
  const int y0  = ty * THT;
  const size_t img_base = (size_t)b * IMG * IMG;

  // Descriptor clipping (uniform per block). DMA never reads out of bounds.
  const int clip_l = (tx == 0)           ? HALOX : 0;
  const int clip_r = (tx == TILES_X - 1) ? HALOX : 0;
  const int clip_t = (ty == 0)           ? HALOY : 0;
  const int clip_b = (ty == TILES_Y - 1) ? HALOY : 0;
  const bool edge  = (clip_l | clip_r | clip_t | clip_b) != 0;

  // Edge tiles: pre-zero LDS so clipped halo cells read as mask=false.
  if (edge) {
    for (int c = tid; c < LH * LWP; c += NTHREADS) s_t[c] = 0.0f;
  }
  __syncthreads();

  // ---- CDNA5 Tensor Data Mover: one DMA moves the whole 2D tile to LDS ----
  const int wid = __builtin_amdgcn_readfirstlane((int)threadIdx.x >> 5);
  if (wid == 0) {
    const int width  = LWP - clip_l - clip_r;      // tile_dim0 (elements)
    const int height = LH  - clip_t - clip_b;      // tile_dim1 (rows)
    const int gx0 = x0 - HALOX + clip_l;
    const int gy0 = y0 - HALOY + clip_t;
    const uint64_t gaddr =
        (uint64_t)(uintptr_t)(targets + img_base + (size_t)gy0 * IMG + gx0);
    const uint32_t laddr =
        (uint32_t)(uintptr_t)(&s_t[clip_t * LWP + clip_l]);  // low 32 = LDS byte addr

    u4 g0;
    g0[0] = 1u;                                      // count=1, user descriptor
    g0[1] = laddr;                                   // lds_addr [63:32]
    g0[2] = (uint32_t)gaddr;                         // global_addr lo
    g0[3] = (uint32_t)(gaddr >> 32) | (2u << 30);    // global_addr hi | type=2

    i8 g1;
    g1[0] = (int)(2u << 16);                         // data_size=4B, wg_mask=0
    g1[1] = (int)((uint32_t)IMG << 16);              // tensor_dim0[15:0] @ bits63:48
    g1[2] = (int)((uint32_t)IMG << 16);              // dim0 hi=0 | tensor_dim1[15:0]
    g1[3] = (int)((uint32_t)width << 16);            // dim1 hi=0 | tile_dim0
    g1[4] = (int)(uint32_t)height;                   // tile_dim1 | tile_dim2=0 (2D)
    g1[5] = (int)IMG;                                // tensor_dim0_stride lo32
    g1[6] = 0;                                       // stride hi | dim1_stride lo
    g1[7] = 0;

    i4 g2 = {0, 0, 0, 0};                            // 2D: groups 2/3 unused
    i4 g3 = {0, 0, 0, 0};

#if __has_builtin(__builtin_amdgcn_tensor_load_to_lds)
#if __clang_major__ >= 23
    i8 gpad = {0, 0, 0, 0, 0, 0, 0, 0};
    __builtin_amdgcn_tensor_load_to_lds(g0, g1, g2, g3, gpad, 0);
#else
    __builtin_amdgcn_tensor_load_to_lds(g0, g1, g2, g3, 0);
#endif
#else
    asm volatile("tensor_load_to_lds %0, %1, %2, %3"
                 :: "s"(g0), "s"(g1), "s"(g2), "s"(g3) : "memory");
#endif
  }
  __builtin_amdgcn_s_wait_tensorcnt(0);
  __syncthreads();

  // Each thread walks a 32-tall column: x = tid&127, rows ysec*32 .. +31.
  const int xloc  = tid & (TW - 1);
  const int lx    = xloc + HALOX;       // 4..131 (stencil reads 2..133)
  const int ysec  = tid >> 7;           // 0 or 1
  const int ybase = ysec * 32;          // LDS row of window top (center-2)

  float acc_focal = 0.0f, acc_inter = 0.0f, acc_p = 0.0f, acc_t = 0.0f, acc_bnd = 0.0f;

  RowAgg r0 = make_row(&s_t[(ybase + 0) * LWP + lx]);
  RowAgg r1 = make_row(&s_t[(ybase + 1) * LWP + lx]);
  RowAgg r2 = make_row(&s_t[(ybase + 2) * LWP + lx]);
  RowAgg r3 = make_row(&s_t[(ybase + 3) * LWP + lx]);

  const float* irow = inputs + img_base + (size_t)(y0 + ysec * 32) * IMG + (x0 + xloc);

  #pragma unroll 4
  for (int j = 0; j < 32; ++j) {
    RowAgg r4 = make_row(&s_t[(ybase + 4 + j) * LWP + lx]);

    // dilated(2x) = OR over diamond(2); eroded(2x) = AND over diamond(2)
    float andv = fminf(fminf(r0.a, r4.a), fminf(fminf(r1.bmin, r3.bmin), r2.cmin));
    float orv  = fmaxf(fmaxf(r0.a, r4.a), fmaxf(fmaxf(r1.bmax, r3.bmax), r2.cmax));
    bool boundary = (orv > 0.5f) && (andv < 0.5f);
    float w = boundary ? 6.0f : 1.0f;   // 1 + THETA*boundary

    float t  = r2.a;                    // center target (0.0 / 1.0)
    float xv = irow[(size_t)j * IMG];

    float e     = __expf(-fabsf(xv));
    float denom = 1.0f + e;
    float rp    = __builtin_amdgcn_rcpf(denom);
    float p     = (xv >= 0.0f) ? rp : e * rp;          // sigmoid(x)
    float bce   = fmaxf(xv, 0.0f) - xv * t + __logf(denom);
    float pt    = fmaf(2.0f * p, t, 1.0f - t - p);     // == exp(-bce) for binary t
    float om    = 1.0f - pt;

    acc_focal = fmaf(0.25f * om * om, bce, acc_focal); // ALPHA=0.25, GAMMA=2
    acc_inter = fmaf(p, t, acc_inter);
    acc_p    += p;
    acc_t    += t;
    acc_bnd   = fmaf(bce, w, acc_bnd);

    r0 = r1; r1 = r2; r2 = r3; r3 = r4;
  }

  // Wave32 reduction, then cross-wave via LDS. Deterministic (no atomics).
  float vals[NACC] = {acc_focal, acc_inter, acc_p, acc_t, acc_bnd};
  const int lane = tid & 31;
  const int wv   = tid >> 5;
  #pragma unroll
  for (int i = 0; i < NACC; ++i) {
    float v = vals[i];
    #pragma unroll
    for (int o = 16; o > 0; o >>= 1) v += __shfl_down(v, o, 32);
    if (lane == 0) s_red[wv][i] = v;
  }
  __syncthreads();
  if (tid == 0) {
    float* pp = partials + (size_t)blk * NACC;
    #pragma unroll
    for (int i = 0; i < NACC; ++i) {
      float s = 0.0f;
      #pragma unroll
      for (int w8 = 0; w8 < 8; ++w8) s += s_red[w8][i];
      pp[i] = s;
    }
  }
}

__global__ __launch_bounds__(256) void loss_final_kernel(
    const float* __restrict__ partials, float* __restrict__ out) {
  __shared__ float sm[NACC][256];
  const int tid = threadIdx.x;
  float s[NACC] = {0.0f, 0.0f, 0.0f, 0.0f, 0.0f};
  for (int i = tid; i < NBLOCKS; i += 256) {
    #pragma unroll
    for (int k = 0; k < NACC; ++k) s[k] += partials[(size_t)i * NACC + k];
  }
  #pragma unroll
  for (int k = 0; k < NACC; ++k) sm[k][tid] = s[k];
  __syncthreads();
  for (int step = 128; step > 0; step >>= 1) {
    if (tid < step) {
      #pragma unroll
      for (int k = 0; k < NACC; ++k) sm[k][tid] += sm[k][tid + step];
    }
    __syncthreads();
  }
  if (tid == 0) {
    const float N = 16777216.0f;  // 64*512*512
    float focal_loss    = sm[0][0] / N;
    float dice          = (2.0f * sm[1][0] + 1e-6f) / (sm[2][0] + sm[3][0] + 1e-6f);
    float boundary_loss = sm[4][0] / N;
    out[0] = 0.3f * focal_loss + 0.4f * (1.0f - dice) + 0.3f * boundary_loss;
  }
}

extern "C" void kernel_launch(void* const* d_in, const int* in_sizes, int n_in,
                              void* d_out, int out_size, void* d_ws, size_t ws_size,
                              hipStream_t stream) {
  (void)in_sizes; (void)n_in; (void)out_size; (void)ws_size;
  const float* inputs  = (const float*)d_in[0];
  const float* targets = (const float*)d_in[1];
  float* partials = (float*)d_ws;   // 2048 * 5 floats = 40 KB
  loss_partial_kernel<<<NBLOCKS, NTHREADS, 0, stream>>>(inputs, targets, partials);
  loss_final_kernel<<<1, 256, 0, stream>>>(partials, (float*)d_out);
}